// TripletAllLoss_50190987821742
// MI455X (gfx1250) — compile-verified
//
#include <hip/hip_runtime.h>
#include <hip/hip_bf16.h>

#define NB 512   // batch
#define ND 512   // feature dim
#define NL 24    // labels

typedef float v2f __attribute__((ext_vector_type(2)));
typedef float v8f __attribute__((ext_vector_type(8)));

// ---------------- row norms: n_i = sum_d S[i][d]^2 ----------------
__global__ void tl_norms_kernel(const float* __restrict__ src, float* __restrict__ norms) {
    int i = blockIdx.x * blockDim.x + threadIdx.x;
    if (i < NB) {
        const float* row = src + (size_t)i * ND;
        float s = 0.f;
        #pragma unroll 4
        for (int d = 0; d < ND; ++d) { float x = row[d]; s = fmaf(x, x, s); }
        norms[i] = s;
    }
}

// ---------------- pack 24 binary labels into a bitmask ----------------
__global__ void tl_mask_kernel(const int* __restrict__ labels, unsigned* __restrict__ masks) {
    int i = blockIdx.x * blockDim.x + threadIdx.x;
    if (i < NB) {
        const int* row = labels + (size_t)i * NL;
        unsigned m = 0u;
        #pragma unroll
        for (int l = 0; l < NL; ++l) m |= (row[l] != 0 ? 1u : 0u) << l;
        masks[i] = m;
    }
}

// ---------------- dist tile via V_WMMA_F32_16X16X4_F32 ----------------
// One wave (32 lanes) computes one 16x16 tile of G = S * S^T, then the
// epilogue converts to dist = sqrt(max(0, n_i + n_j - 2*G)) with the
// reference's zero handling (dist==0 -> 0 exactly).
__global__ void tl_dist_wmma_kernel(const float* __restrict__ src,
                                    const float* __restrict__ norms,
                                    float* __restrict__ dist) {
    const int row0 = (blockIdx.y * blockDim.y + threadIdx.y) * 16;
    const int col0 = blockIdx.x * 16;
    const int lane = threadIdx.x & 31;
    const int h    = lane >> 4;   // half-wave select (K pair for A/B frags)
    const int m    = lane & 15;   // row (A) / col (B) index within tile

    // A frag: A[m][k + 2h + {0,1}]  (16x4 f32, ISA layout)
    // B frag: B[k'][n] = S[col0+n][k'] -> per-lane contiguous pair from row col0+m
    const float* arow = src + (size_t)(row0 + m) * ND + 2 * h;
    const float* brow = src + (size_t)(col0 + m) * ND + 2 * h;

    v8f acc = {};
    #pragma unroll 8
    for (int k = 0; k < ND; k += 4) {
        v2f a = *(const v2f*)(arow + k);
        v2f b = *(const v2f*)(brow + k);
        // 8 args: (neg_a, A, neg_b, B, c_mod, C, reuse_a, reuse_b)
        acc = __builtin_amdgcn_wmma_f32_16x16x4_f32(false, a, false, b,
                                                    (short)0, acc, false, false);
    }

    const float nj = norms[col0 + m];
    #pragma unroll
    for (int v = 0; v < 8; ++v) {
        int r = row0 + v + 8 * h;          // C/D layout: VGPR v, half h -> row
        float d2 = norms[r] + nj - 2.0f * acc[v];
        d2 = fmaxf(d2, 0.0f);
        dist[(size_t)r * NB + col0 + m] = (d2 > 0.0f) ? __builtin_sqrtf(d2) : 0.0f;
    }
}

// ---------------- zero the accumulators ----------------
__global__ void tl_init_kernel(float* __restrict__ accum) {
    if (threadIdx.x < 2) accum[threadIdx.x] = 0.0f;
}

// ---------------- B^3 triplet reduction, one anchor i per block ----------------
__global__ void tl_triplet_kernel(const float* __restrict__ dist,
                                  const unsigned* __restrict__ masks,
                                  float* __restrict__ accum) {
    __shared__ float    drow[NB];
    __shared__ unsigned sflag[NB];
    __shared__ float    rs[256];
    __shared__ float    rc[256];

    const int i = blockIdx.x;
    const unsigned mi = masks[i];
    for (int j = threadIdx.x; j < NB; j += 256) {
        drow[j]  = dist[(size_t)i * NB + j];
        sflag[j] = (mi & masks[j]) ? 1u : 0u;   // sim(i,j)
    }
    __syncthreads();

    float sum = 0.0f, cnt = 0.0f;
    for (int j = threadIdx.x; j < NB; j += 256) {
        if (!sflag[j]) continue;                 // need sim(i,j)==1
        const float dij = drow[j];
        #pragma unroll 4
        for (int k = 0; k < NB; ++k) {
            if (sflag[k]) continue;              // need sim(i,k)==0
            float v = dij - drow[k];
            if (v > 0.0f)      sum += v;         // sum of clip(triplet, 0)
            if (v > 1e-16f)    cnt += 1.0f;      // num_positive
        }
    }

    rs[threadIdx.x] = sum;
    rc[threadIdx.x] = cnt;
    __syncthreads();
    for (int s = 128; s > 0; s >>= 1) {
        if (threadIdx.x < s) {
            rs[threadIdx.x] += rs[threadIdx.x + s];
            rc[threadIdx.x] += rc[threadIdx.x + s];
        }
        __syncthreads();
    }
    if (threadIdx.x == 0) {
        atomicAdd(&accum[0], rs[0]);
        atomicAdd(&accum[1], rc[0]);
    }
}

// ---------------- loss = sum / (count + 1e-16) ----------------
__global__ void tl_final_kernel(const float* __restrict__ accum, float* __restrict__ out) {
    if (threadIdx.x == 0) out[0] = accum[0] / (accum[1] + 1e-16f);
}

extern "C" void kernel_launch(void* const* d_in, const int* in_sizes, int n_in,
                              void* d_out, int out_size, void* d_ws, size_t ws_size,
                              hipStream_t stream) {
    const float* src    = (const float*)d_in[0];   // (512, 512) f32
    const int*   labels = (const int*)d_in[1];     // (512, 24) i32
    float*       out    = (float*)d_out;           // scalar f32

    // workspace layout
    char* ws = (char*)d_ws;
    float*    norms = (float*)(ws);                         // 512 * 4 = 2 KB
    unsigned* masks = (unsigned*)(ws + 4096);               // 512 * 4 = 2 KB
    float*    accum = (float*)(ws + 8192);                  // 2 floats
    float*    dist  = (float*)(ws + 16384);                 // 512*512*4 = 1 MB

    tl_norms_kernel<<<dim3(2), dim3(256), 0, stream>>>(src, norms);
    tl_mask_kernel <<<dim3(2), dim3(256), 0, stream>>>(labels, masks);

    // 32x32 tiles of 16x16; 4 waves per block, one tile each
    tl_dist_wmma_kernel<<<dim3(NB / 16, NB / 16 / 4), dim3(32, 4), 0, stream>>>(src, norms, dist);

    tl_init_kernel   <<<dim3(1),  dim3(32),  0, stream>>>(accum);
    tl_triplet_kernel<<<dim3(NB), dim3(256), 0, stream>>>(dist, masks, accum);
    tl_final_kernel  <<<dim3(1),  dim3(32),  0, stream>>>(accum, out);
}